// NodeGLAM_60979945669285
// MI455X (gfx1250) — compile-verified
//
#include <hip/hip_runtime.h>
#include <math.h>

typedef float v2f __attribute__((ext_vector_type(2)));
typedef float v8f __attribute__((ext_vector_type(8)));

__device__ __forceinline__ float gelu_f(float x) {
    return 0.5f * x * (1.0f + erff(x * 0.70710678118654752f));
}

// ---------------- zero fill (float4) ----------------
__global__ void k_zero(float* __restrict__ p, long long n) {
    long long i = (long long)blockIdx.x * blockDim.x + threadIdx.x;
    long long stride = (long long)gridDim.x * blockDim.x;
    long long n4 = n >> 2;
    float4 z = make_float4(0.f, 0.f, 0.f, 0.f);
    for (; i < n4; i += stride) ((float4*)p)[i] = z;
}

// ---------------- column stats: sum, sumsq over N rows of x[N,128] ----------------
__global__ void k_colstats(const float* __restrict__ x, float* __restrict__ sum,
                           float* __restrict__ sq, int N) {
    int c = threadIdx.x;  // 128 threads = 128 columns
    float s = 0.f, s2 = 0.f;
    for (int i = blockIdx.x; i < N; i += gridDim.x) {
        float v = x[(size_t)i * 128 + c];
        s += v; s2 += v * v;
    }
    atomicAdd(&sum[c], s);
    atomicAdd(&sq[c], s2);
}

__global__ void k_finalize_stats(const float* __restrict__ sum, const float* __restrict__ sq,
                                 const float* __restrict__ gamma, const float* __restrict__ beta,
                                 float* __restrict__ scale, float* __restrict__ shift, int N) {
    int c = threadIdx.x;
    float inv_n = 1.0f / (float)N;
    float mean = sum[c] * inv_n;
    float var = sq[c] * inv_n - mean * mean;
    float s = gamma[c] * rsqrtf(var + 1e-5f);
    scale[c] = s;
    shift[c] = beta[c] - mean * s;
}

// ---------------- degree + normalization ----------------
__global__ void k_deg(const int* __restrict__ col, float* __restrict__ deg, int E) {
    int e = blockIdx.x * blockDim.x + threadIdx.x;
    if (e < E) atomicAdd(&deg[col[e]], 1.0f);
}

__global__ void k_dis(const float* __restrict__ deg, float* __restrict__ dis, int N) {
    int i = blockIdx.x * blockDim.x + threadIdx.x;
    if (i < N) {
        float d = deg[i];
        dis[i] = (d > 0.f) ? rsqrtf(fmaxf(d, 1.0f)) : 0.0f;
    }
}

// ---------------- xb = x*scale + shift (float4 vectorized) ----------------
__global__ void k_affine(const float* __restrict__ x, const float* __restrict__ scale,
                         const float* __restrict__ shift, float* __restrict__ xb, long long n4) {
    long long i = (long long)blockIdx.x * blockDim.x + threadIdx.x;
    if (i >= n4) return;
    int cb = (int)(i & 31) * 4;  // column base within 128
    float4 v = ((const float4*)x)[i];
    float4 sc = *(const float4*)(scale + cb);
    float4 sh = *(const float4*)(shift + cb);
    float4 r;
    r.x = v.x * sc.x + sh.x; r.y = v.y * sc.y + sh.y;
    r.z = v.z * sc.z + sh.z; r.w = v.w * sc.w + sh.w;
    ((float4*)xb)[i] = r;
}

// ---------------- generic WMMA f32 GEMM: C[M,128] (+)= op(A[M,128]) @ B[128,128] ----------------
// Template flags keep the hot loop branch-free: GIN (gelu on A load), ACC (C += result),
// HASB (add bias), GOUT (gelu on store).
// B staged in LDS as row-pairs (float2) so each WMMA B-fragment is a single aligned
// 8-byte ds load into an even VGPR pair; column swizzled by ((rowpair&1)<<4) so the two
// half-waves (row-pairs p and p+1) hit disjoint 32-bank halves.
template <int GIN, int ACC, int HASB, int GOUT>
__global__ __launch_bounds__(256) void k_gemm128(
    const float* __restrict__ A, const float* __restrict__ B, float* __restrict__ C,
    const float* __restrict__ bias, int M)
{
    __shared__ float Bs[128 * 128];  // 64 KB
    for (int i = threadIdx.x; i < 128 * 128; i += 256) {
        int r = i >> 7, c = i & 127;
        int p = r >> 1;
        int idx = (p * 128 + (c ^ ((p & 1) << 4))) * 2 + (r & 1);
        Bs[idx] = B[i];
    }
    __syncthreads();

    const int lane = threadIdx.x & 31;
    const int wave = threadIdx.x >> 5;
    const int half = lane >> 4;   // 0: lanes 0-15, 1: lanes 16-31
    const int ln   = lane & 15;   // M index for A operand, N index for B/C
    const int tiles = M >> 4;
    // per-lane LDS base: row-pair (k/2 + half), column ln (swizzle folded in)
    const float* Bbase = Bs + half * 256 + 2 * ln;

    for (int t = blockIdx.x * 8 + wave; t < tiles; t += gridDim.x * 8) {
        const float* Arow = A + (size_t)(t * 16 + ln) * 128 + 2 * half;
        v8f acc[8];
#pragma unroll
        for (int nt = 0; nt < 8; ++nt) acc[nt] = (v8f){0,0,0,0,0,0,0,0};

#pragma unroll 2
        for (int k = 0; k < 128; k += 4) {
            float ax = Arow[k];
            float ay = Arow[k + 1];
            if (GIN) { ax = gelu_f(ax); ay = gelu_f(ay); }
            v2f a2 = {ax, ay};
            const float* bp = Bbase + (k >> 1) * 256;
#pragma unroll
            for (int nt = 0; nt < 8; ++nt) {
                v2f b2 = *(const v2f*)(bp + ((nt ^ half) << 5));  // fragment for col-group nt
                acc[nt] = __builtin_amdgcn_wmma_f32_16x16x4_f32(
                    false, a2, false, b2, (short)0, acc[nt], false, false);
            }
        }

        const int crow0 = t * 16 + 8 * half;  // acc[nt][r] -> row crow0+r
#pragma unroll
        for (int nt = 0; nt < 8; ++nt) {
#pragma unroll
            for (int r = 0; r < 8; ++r) {
                size_t idx = (size_t)(crow0 + r) * 128 + nt * 16 + ln;
                float v = acc[nt][r];
                if (ACC)  v += C[idx];
                if (HASB) v += bias[nt * 16 + ln];
                if (GOUT) v = gelu_f(v);
                C[idx] = v;
            }
        }
    }
}

// ---------------- one hop: xnew[col] += dis[row]*dis[col] * xold[row]  (wave per edge) ------
__global__ void k_scatter(const float* __restrict__ xold, float* __restrict__ xnew,
                          const int* __restrict__ row, const int* __restrict__ col,
                          const float* __restrict__ dis, int E)
{
    long long t = (long long)blockIdx.x * blockDim.x + threadIdx.x;
    int lane = (int)(t & 31);
    long long e = t >> 5;
    if (e >= E) return;
    int r = row[e], c = col[e];
    float w = dis[r] * dis[c];
    float4 v = *(const float4*)(xold + (size_t)r * 128 + lane * 4);
    float* dst = xnew + (size_t)c * 128 + lane * 4;
    atomicAdd(dst + 0, v.x * w);
    atomicAdd(dst + 1, v.y * w);
    atomicAdd(dst + 2, v.z * w);
    atomicAdd(dst + 3, v.w * w);
}

// ---------------- elementwise gelu (float4) ----------------
__global__ void k_gelu_ew(const float* __restrict__ in, float* __restrict__ out, long long n4) {
    long long i = (long long)blockIdx.x * blockDim.x + threadIdx.x;
    if (i >= n4) return;
    float4 v = ((const float4*)in)[i];
    float4 r;
    r.x = gelu_f(v.x); r.y = gelu_f(v.y); r.z = gelu_f(v.z); r.w = gelu_f(v.w);
    ((float4*)out)[i] = r;
}

// ---------------- classifier: cl[N,10] = ga[N,128] @ Wc[128,10] + bc ----------------
__global__ void k_cls(const float* __restrict__ ga, const float* __restrict__ Wc,
                      const float* __restrict__ bc, float* __restrict__ out, int N)
{
    long long t = (long long)blockIdx.x * blockDim.x + threadIdx.x;
    int c = (int)(t & 15);
    long long i = t >> 4;
    if (i >= N || c >= 10) return;
    float acc = bc[c];
    const float* g = ga + i * 128;
#pragma unroll 8
    for (int k = 0; k < 128; ++k) acc += g[k] * Wc[k * 10 + c];
    out[i * 10 + c] = acc;
}

extern "C" void kernel_launch(void* const* d_in, const int* in_sizes, int n_in,
                              void* d_out, int out_size, void* d_ws, size_t ws_size,
                              hipStream_t stream) {
    const float* x     = (const float*)d_in[0];
    const int*   ei    = (const int*)  d_in[1];
    const float* gamma = (const float*)d_in[2];
    const float* beta  = (const float*)d_in[3];
    const float* Wt    = (const float*)d_in[4];
    const float* bt    = (const float*)d_in[5];
    const float* tagW  = (const float*)d_in[6];
    const float* tagb  = (const float*)d_in[7];
    const float* W1    = (const float*)d_in[8];
    const float* b1    = (const float*)d_in[9];
    const float* Wc    = (const float*)d_in[10];
    const float* bc    = (const float*)d_in[11];

    const int N = in_sizes[0] / 128;
    const int E = in_sizes[1] / 2;
    const size_t NF = (size_t)N * 128;

    float* ws    = (float*)d_ws;
    float* xb    = ws;            // [N,128]
    float* bufA  = ws + NF;       // h / xk ping
    float* bufB  = ws + 2 * NF;   // xk pong
    float* outA  = ws + 3 * NF;   // TAGConv accumulator -> h2
    float* deg   = ws + 4 * NF;   // [N]
    float* stats = deg + N;       // sum[128] sq[128] scale[128] shift[128]
    float* sum   = stats, *sq = stats + 128, *scale = stats + 256, *shift = stats + 384;
    float* dis   = stats + 512;   // [N]

    float* a_out  = (float*)d_out;       // [N,128]
    float* cl_out = a_out + NF;          // [N,10]

    const int* erow = ei;
    const int* ecol = ei + E;

    const int GG = (N / 16 + 7) / 8;     // waves-per-block = 8, one 16-row tile per wave

    // ---- stats & graph norm ----
    k_zero<<<1024, 256, 0, stream>>>(deg, (long long)(N + 256));  // deg + sum + sq
    k_colstats<<<512, 128, 0, stream>>>(x, sum, sq, N);
    k_deg<<<(E + 255) / 256, 256, 0, stream>>>(ecol, deg, E);
    k_finalize_stats<<<1, 128, 0, stream>>>(sum, sq, gamma, beta, scale, shift, N);
    k_dis<<<(N + 255) / 256, 256, 0, stream>>>(deg, dis, N);

    // ---- xb = BN(x) ----
    long long n4 = (long long)NF / 4;
    k_affine<<<(int)((n4 + 255) / 256), 256, 0, stream>>>(x, scale, shift, xb, n4);

    // ---- h = gelu(xb @ Wt + bt) ----
    k_gemm128<0,0,1,1><<<GG, 256, 0, stream>>>(xb, Wt, bufA, bt, N);
    // ---- out = h @ W0 ----
    k_gemm128<0,0,0,0><<<GG, 256, 0, stream>>>(bufA, tagW, outA, nullptr, N);

    // ---- K=6 propagation hops, out += xk @ Wk; last hop fuses +tag_b, gelu ----
    for (int k = 1; k <= 6; ++k) {
        float* cur = (k & 1) ? bufA : bufB;
        float* nxt = (k & 1) ? bufB : bufA;
        k_zero<<<4096, 256, 0, stream>>>(nxt, (long long)NF);
        long long st = (long long)E * 32;
        k_scatter<<<(int)((st + 255) / 256), 256, 0, stream>>>(cur, nxt, erow, ecol, dis, E);
        const float* Bk = tagW + (size_t)k * 128 * 128;
        if (k < 6)
            k_gemm128<0,1,0,0><<<GG, 256, 0, stream>>>(nxt, Bk, outA, nullptr, N);
        else  // h2 = gelu(out + xk6@W6 + tag_b), stored back into outA
            k_gemm128<0,1,1,1><<<GG, 256, 0, stream>>>(nxt, Bk, outA, tagb, N);
    }

    // ---- a = gelu(xb)@W1[0:128] + gelu(h2)@W1[128:256] + b1 ----
    k_gemm128<1,0,1,0><<<GG, 256, 0, stream>>>(xb,   W1,             a_out, b1,      N);
    k_gemm128<1,1,0,0><<<GG, 256, 0, stream>>>(outA, W1 + 128 * 128, a_out, nullptr, N);

    // ---- cl = gelu(a) @ Wc + bc ----
    k_gelu_ew<<<(int)((n4 + 255) / 256), 256, 0, stream>>>(a_out, bufA, n4);
    long long ct = (long long)N * 16;
    k_cls<<<(int)((ct + 255) / 256), 256, 0, stream>>>(bufA, Wc, bc, cl_out, N);
}